// ConvQuadInterp3d_63453846831185
// MI455X (gfx1250) — compile-verified
//
#include <hip/hip_runtime.h>
#include <stdint.h>

// Problem constants (match reference)
#define Bn 4
#define Dn 4
#define Hn 768
#define Wn 768
#define BONUS 10.0f

// Tiling: 256 threads = 8 wave32s; each block does an 8x32 (H x W) tile,
// all 4 depth planes of one batch image.
#define TH 8
#define TW 32
#define HALO_H (TH + 2)
#define HALO_W (TW + 2)
#define NELEM  (Dn * HALO_H * HALO_W)   // 1360 floats = 5.4 KB LDS
#define NTHREADS 256

typedef __attribute__((address_space(3))) float lds_f;

__global__ __launch_bounds__(NTHREADS)
void conv_quad_interp3d_kernel(const float* __restrict__ x, float* __restrict__ out)
{
    __shared__ float smem[NELEM];  // [dp][lh][lw]

    const int tid = threadIdx.x;
    const int tx  = tid & (TW - 1);     // w within tile
    const int ty  = tid >> 5;           // h within tile
    const int b   = blockIdx.z;
    const int h0  = blockIdx.y * TH;
    const int w0  = blockIdx.x * TW;

    // ---------------------------------------------------------------
    // Phase 1: halo load, global -> LDS via CDNA5 async copy engine.
    // Per-lane clamped addresses implement replicate padding in H/W.
    // Tracked by ASYNCcnt; no VGPR staging of the data.
    // ---------------------------------------------------------------
    #pragma unroll
    for (int i0 = 0; i0 < NELEM; i0 += NTHREADS) {
        int i = i0 + tid;
        if (i < NELEM) {
            int dp = i / (HALO_H * HALO_W);
            int r  = i - dp * (HALO_H * HALO_W);
            int lh = r / HALO_W;
            int lw = r - lh * HALO_W;
            int gh = h0 - 1 + lh; gh = gh < 0 ? 0 : (gh > Hn - 1 ? Hn - 1 : gh);
            int gw = w0 - 1 + lw; gw = gw < 0 ? 0 : (gw > Wn - 1 ? Wn - 1 : gw);
            const float* src = x + (((size_t)b * Dn + dp) * Hn + gh) * (size_t)Wn + gw;
            unsigned lds_off = (unsigned)(size_t)(lds_f*)(&smem[i]);  // 32-bit LDS byte offset
            asm volatile("global_load_async_to_lds_b32 %0, %1, off"
                         :
                         : "v"(lds_off), "v"(src)
                         : "memory");
        }
    }
    asm volatile("s_wait_asynccnt 0" ::: "memory");  // ASYNCcnt == 0 for this wave
    __syncthreads();                                 // publish LDS across all 8 waves

    // ---------------------------------------------------------------
    // Phase 2: pull the four 3x3 plane neighborhoods into registers.
    // Depth replicate padding is handled by plane-pointer reuse below.
    // ---------------------------------------------------------------
    float q[Dn][9];
    #pragma unroll
    for (int dp = 0; dp < Dn; ++dp)
        #pragma unroll
        for (int dh = 0; dh < 3; ++dh)
            #pragma unroll
            for (int dw = 0; dw < 3; ++dw)
                q[dp][dh * 3 + dw] = smem[(dp * HALO_H + ty + dh) * HALO_W + (tx + dw)];

    const size_t HW  = (size_t)Hn * Wn;
    const size_t V   = (size_t)Dn * HW;              // per-(b,channel) volume
    const size_t pix = (size_t)(h0 + ty) * Wn + (w0 + tx);
    const float  fw  = (float)(w0 + tx);
    const float  fh  = (float)(h0 + ty);

    // ---------------------------------------------------------------
    // Phase 3: per-depth stencil math + Cramer solve + stores.
    // ---------------------------------------------------------------
    #pragma unroll
    for (int d = 0; d < Dn; ++d) {
        const float* m = q[d == 0 ? 0 : d - 1];       // depth replicate pad
        const float* c = q[d];
        const float* p = q[d == Dn - 1 ? Dn - 1 : d + 1];

        const float xc = c[4];

        // first derivatives
        const float gx = 0.5f * (c[5] - c[3]);
        const float gy = 0.5f * (c[7] - c[1]);
        const float gs = 0.5f * (p[4] - m[4]);
        // second derivatives
        const float dxx = c[5] - 2.0f * xc + c[3];
        const float dyy = c[7] - 2.0f * xc + c[1];
        const float dss = p[4] - 2.0f * xc + m[4];
        // cross terms (with kornia's sign flips on ys/xs)
        const float dxy =  0.25f * (c[8] - c[6] - c[2] + c[0]);
        const float dys = -0.25f * (p[7] - p[1] - m[7] + m[1]);
        const float dxs = -0.25f * (p[5] - p[3] - m[5] + m[3]);

        // strict NMS over 26 neighbors, seeded with 0 (=> also requires x > 0).
        // At image borders a clamped neighbor equals xc so the strict > fails,
        // exactly matching the replicate-padded reference.
        float mx = 0.0f;
        #pragma unroll
        for (int k = 0; k < 9; ++k) {
            mx = fmaxf(mx, m[k]);
            mx = fmaxf(mx, p[k]);
            if (k != 4) mx = fmaxf(mx, c[k]);
        }
        const bool nms = xc > mx;

        // symmetric 3x3 Cramer solve H s = g
        const float cf00 = dyy * dss - dys * dys;
        const float cf01 = dxy * dss - dys * dxs;
        const float cf02 = dxy * dys - dyy * dxs;
        const float det  = dxx * cf00 - dxy * cf01 + dxs * cf02;
        const bool  solved = (det != 0.0f);           // |det| > 0
        const float safe   = solved ? det : 1.0f;
        const float inv    = __builtin_amdgcn_rcpf(safe);

        const float t0 = gy * dss - dys * gs;
        const float t1 = gy * dys - dyy * gs;
        const float t2 = dxy * gs - gy * dxs;
        const float sx = (gx * cf00 - dxy * t0 + dxs * t1) * inv;
        const float sy = (dxx * t0 - gx * cf01 + dxs * t2) * inv;
        const float ss = (dxx * (dyy * gs - gy * dys) - dxy * t2 + gx * cf02) * inv;

        const bool valid = nms && solved;
        float dxv = valid ? -sx : 0.0f;
        float dyv = valid ? -sy : 0.0f;
        float dsv = valid ? -ss : 0.0f;

        const bool big = fmaxf(fmaxf(fabsf(dxv), fabsf(dyv)), fabsf(dsv)) > 0.7f;
        if (big) { dxv = 0.0f; dyv = 0.0f; dsv = 0.0f; }

        const float dy2  = 0.5f * (gx * dxv + gy * dyv + gs * dsv);
        const float ymax = xc + dy2 + (valid ? BONUS : 0.0f);

        // coords_max (B,C,3,D,H,W), channel order (z, x, y); then y_max (B,C,D,H,W)
        const size_t cbase = (size_t)b * 3 * V + (size_t)d * HW + pix;
        out[cbase]          = (float)d + dsv;   // ch0: depth coord
        out[cbase + V]      = fw + dxv;         // ch1: x coord
        out[cbase + 2 * V]  = fh + dyv;         // ch2: y coord
        out[(size_t)Bn * 3 * V + (size_t)b * V + (size_t)d * HW + pix] = ymax;
    }
}

extern "C" void kernel_launch(void* const* d_in, const int* in_sizes, int n_in,
                              void* d_out, int out_size, void* d_ws, size_t ws_size,
                              hipStream_t stream) {
    const float* x = (const float*)d_in[0];
    float* out = (float*)d_out;
    dim3 grid(Wn / TW, Hn / TH, Bn);   // 24 x 96 x 4 = 9216 blocks
    conv_quad_interp3d_kernel<<<grid, NTHREADS, 0, stream>>>(x, out);
}